// CostVolume3D_22531398435046
// MI455X (gfx1250) — compile-verified
//
#include <hip/hip_runtime.h>
#include <cstdint>

// CostVolume3D for MI455X (gfx1250, wave32).
//
// Derivation: every tile(...,D)+row-major-reshape pair maps tiled flat index f
// to the untiled tensor at f/23. Per output flat index g:
//   row = g/256, w' = g%256, s = (row/128)%23 - 11
//   disp = wflow[g/23] - s ; xq = w' - disp ; x0 = floor(xq) ; fr = xq - x0
//   x0i = clip(int(x0),0,255) ; x1i = clip(x0i+1,0,255)
//   out[g] = sum_{c<8} | L_c - (R0_c*(1-fr) + R1_c*fr) |
//     L_c  = feat_l[(g*8+c)/23]
//     R0_c = feat_r[((row*256+x0i)*8+c)/23],  R1_c likewise with x1i
// For each source the 8 channel indices span at most 2 consecutive slots,
// switching at channel threshold t = clamp(23 - (base%23), 0, 8). So the
// per-channel difference is piecewise constant with <=4 runs; we sum
// run_len * |run_value| instead of an 8-iteration select loop (~40% VALU cut;
// the kernel is VALU-bound: ~600M lane-ops vs ~2us memory floor).
//
// Memory plan: one block per row. A row only touches feat indices
// [row*2048/23, (row*2048+2047)/23] (<=91 floats each for L and R) and
// wflow indices [row*256/23, (row*256+255)/23] (<=13 floats). Stage those into
// LDS with GLOBAL_LOAD_ASYNC_TO_LDS_B32 (ASYNCcnt), s_wait_asynccnt, barrier,
// then all gathers hit LDS. ~41 MB total HBM traffic at 23.3 TB/s.

namespace {
constexpr uint32_t kD    = 23;                 // disparity hypotheses
constexpr uint32_t kW    = 256;                // width
constexpr uint32_t kSpan = 2048;               // W*C flat tiled span per row
constexpr uint32_t kRows = 8u * 23u * 128u;    // B*D*H = 23552
constexpr uint32_t kCapA = 92;                 // >= max staged feat floats/row
constexpr uint32_t kCapW = 14;                 // >= max staged wflow floats/row
}

__global__ __launch_bounds__(256) void cost_volume_kernel(
    const float* __restrict__ feat_l,
    const float* __restrict__ feat_r,
    const float* __restrict__ wflow,
    float* __restrict__ out)
{
  __shared__ float sL[kCapA];
  __shared__ float sR[kCapA];
  __shared__ float sW[kCapW];

  const uint32_t row   = blockIdx.x;
  const uint32_t tid   = threadIdx.x;
  const uint32_t g0    = row * kW;          // first output index of this row
  const uint32_t abase = row * kSpan;       // first tiled-feat flat index
  const uint32_t qlo   = abase / kD;
  const uint32_t qhi   = (abase + kSpan - 1u) / kD;
  const uint32_t cntA  = qhi - qlo + 1u;    // <= 91
  const uint32_t wlo   = g0 / kD;
  const uint32_t whi   = (g0 + kW - 1u) / kD;
  const uint32_t cntW  = whi - wlo + 1u;    // <= 13

  // ---- stage this row's working set into LDS with async global->LDS copies.
  // Three disjoint thread ranges; the SGPR base operand is uniform per branch.
  if (tid < cntA) {
    const uint32_t lds  = (uint32_t)(uintptr_t)(sL + tid);
    const uint32_t boff = tid * 4u;
    asm volatile("global_load_async_to_lds_b32 %0, %1, %2"
                 :: "v"(lds), "v"(boff), "s"(feat_l + qlo) : "memory");
  } else if (tid < 2u * cntA) {
    const uint32_t li   = tid - cntA;
    const uint32_t lds  = (uint32_t)(uintptr_t)(sR + li);
    const uint32_t boff = li * 4u;
    asm volatile("global_load_async_to_lds_b32 %0, %1, %2"
                 :: "v"(lds), "v"(boff), "s"(feat_r + qlo) : "memory");
  } else if (tid < 2u * cntA + cntW) {
    const uint32_t li   = tid - 2u * cntA;
    const uint32_t lds  = (uint32_t)(uintptr_t)(sW + li);
    const uint32_t boff = li * 4u;
    asm volatile("global_load_async_to_lds_b32 %0, %1, %2"
                 :: "v"(lds), "v"(boff), "s"(wflow + wlo) : "memory");
  }
  asm volatile("s_wait_asynccnt 0" ::: "memory");
  __syncthreads();

  // ---- per-output compute (one output per lane, coalesced store)
  const uint32_t g  = g0 + tid;
  const int      s  = (int)((row >> 7) % kD) - 11;   // shift, uniform per row
  const float    sf = (float)s;

  const float wv   = sW[g / kD - wlo];
  const float disp = wv - sf;
  const float xq   = (float)tid - disp;
  const float x0f  = floorf(xq);
  const float fr   = xq - x0f;
  int x0i = (int)x0f;
  x0i = min(max(x0i, 0), 255);
  const int x1i = min(x0i + 1, 255);

  // feat_l: two candidate slots + channel-switch threshold
  const uint32_t aL = g * 8u;
  const uint32_t qL = aL / kD;
  const int      tL = 23 - (int)(aL - qL * kD);      // c < tL -> slot qL
  const uint32_t iL = qL - qlo;
  const float lA = sL[iL];
  const float lB = sL[min(iL + 1u, cntA - 1u)];

  // feat_r at x0i
  const uint32_t aR0 = abase + (uint32_t)x0i * 8u;
  const uint32_t qR0 = aR0 / kD;
  const int      tR0 = 23 - (int)(aR0 - qR0 * kD);
  const uint32_t iR0 = qR0 - qlo;
  const float r0A = sR[iR0];
  const float r0B = sR[min(iR0 + 1u, cntA - 1u)];

  // feat_r at x1i
  const uint32_t aR1 = abase + (uint32_t)x1i * 8u;
  const uint32_t qR1 = aR1 / kD;
  const int      tR1 = 23 - (int)(aR1 - qR1 * kD);
  const uint32_t iR1 = qR1 - qlo;
  const float r1A = sR[iR1];
  const float r1B = sR[min(iR1 + 1u, cntA - 1u)];

  const float omf = 1.0f - fr;

  // Piecewise-constant channel difference:
  //   d(c) = dA + [c>=u]*du + [c>=v]*dv + [c>=w]*dw
  // with per-source switch thresholds clamped to [0,8] and deltas:
  int u = min(tL, 8);                 // L switches lA -> lB at c == u
  int v = min(tR0, 8);                // R0 switches at c == v
  int w = min(tR1, 8);                // R1 switches at c == w
  float du = lB - lA;
  float dv = (r0A - r0B) * omf;       // d loses r0A*omf, gains r0B*omf, negated
  float dw = (r1A - r1B) * fr;
  const float dA = lA - (r0A * omf + r1A * fr);

  // Sort the 3 (threshold, delta) pairs ascending: network (0,1),(1,2),(0,1).
  {
    bool c01 = u > v;
    int  tu = c01 ? v : u;  int  tv = c01 ? u : v;
    float su = c01 ? dv : du; float sv = c01 ? du : dv;
    u = tu; v = tv; du = su; dv = sv;
  }
  {
    bool c12 = v > w;
    int  tv = c12 ? w : v;  int  tw = c12 ? v : w;
    float sv = c12 ? dw : dv; float sw = c12 ? dv : dw;
    v = tv; w = tw; dv = sv; dw = sw;
  }
  {
    bool c01 = u > v;
    int  tu = c01 ? v : u;  int  tv = c01 ? u : v;
    float su = c01 ? dv : du; float sv = c01 ? du : dv;
    u = tu; v = tv; du = su; dv = sv;
  }

  // Run values and lengths (u <= v <= w after sorting, all in [0,8]).
  const float s0 = dA;
  const float s1 = s0 + du;
  const float s2 = s1 + dv;
  const float s3 = s2 + dw;
  const float sum = (float)u       * fabsf(s0)
                  + (float)(v - u) * fabsf(s1)
                  + (float)(w - v) * fabsf(s2)
                  + (float)(8 - w) * fabsf(s3);

  out[g] = sum;
}

extern "C" void kernel_launch(void* const* d_in, const int* in_sizes, int n_in,
                              void* d_out, int out_size, void* d_ws, size_t ws_size,
                              hipStream_t stream) {
  (void)in_sizes; (void)n_in; (void)out_size; (void)d_ws; (void)ws_size;
  const float* feat_l = (const float*)d_in[0];
  const float* feat_r = (const float*)d_in[1];
  const float* wflow  = (const float*)d_in[2];
  float* out = (float*)d_out;
  hipLaunchKernelGGL(cost_volume_kernel, dim3(kRows), dim3(256), 0, stream,
                     feat_l, feat_r, wflow, out);
}